// Net_83013127897374
// MI455X (gfx1250) — compile-verified
//
#include <hip/hip_runtime.h>
#include <hip/hip_bf16.h>

typedef float v2f __attribute__((ext_vector_type(2)));
typedef float v8f __attribute__((ext_vector_type(8)));

#define BM 64
#define BN 64
#define BK 16
#define A_PAD 20   // A tile row stride in floats ([m][k] layout, padded: 16B-aligned rows, conflict-free)
#define B_PAD 80   // B tile row stride in floats ([k][n] layout, padded: disjoint bank ranges for half-waves)

// C[M,N] = act(A[M,K] @ Bw[K,N] + bias[N]); C row stride ldC (for concat writes).
// If sel != nullptr, Bw += sel[0]*selStrideW and bias += sel[0]*selStrideB (device-side domain routing).
// ACT: 0 = identity, 1 = relu, 2 = leaky(0.001)
template<int ACT>
__global__ __launch_bounds__(256) void gemm_wmma_f32(
    const float* __restrict__ A, const float* __restrict__ Bw,
    const float* __restrict__ bias, float* __restrict__ C,
    int M, int N, int K, int ldC,
    const int* __restrict__ sel, long selStrideW, long selStrideB)
{
    if (sel) { long s = (long)sel[0]; Bw += s * selStrideW; bias += s * selStrideB; }

    __shared__ float As[2][BM * A_PAD];
    __shared__ float Bs[2][BK * B_PAD];

    const int tid  = threadIdx.x;
    const int lane = tid & 31;
    const int wid  = tid >> 5;      // 0..7 (8 waves, wave32)
    const int mw   = wid & 3;       // wave's 16-row strip within 64
    const int nw   = wid >> 2;      // wave's 32-col strip within 64

    const int m0 = blockIdx.y * BM;
    const int n0 = blockIdx.x * BN;

    // global -> LDS staging map (one float4 per thread per tile)
    const int am = tid >> 2;            // 0..63 : A tile row
    const int ak = (tid & 3) * 4;       // 0/4/8/12 : A tile k
    const int bk = tid >> 4;            // 0..15 : B tile k-row
    const int bn = (tid & 15) * 4;      // 0..60 : B tile col

    const float* Aptr = A  + (long)(m0 + am) * K + ak;
    const float* Bptr = Bw + (long)bk * N + (n0 + bn);

    v8f acc0 = {}; v8f acc1 = {};

    // prologue: stage first k-tile
    float4 aReg = *(const float4*)(Aptr);
    float4 bReg = *(const float4*)(Bptr);
    *(float4*)&As[0][am * A_PAD + ak] = aReg;
    *(float4*)&Bs[0][bk * B_PAD + bn] = bReg;
    __syncthreads();

    // fragment indices (ISA 7.12.2: lanes 0-15 hold K=0,1; lanes 16-31 hold K=2,3)
    const int koff  = (lane >> 4) * 2;
    const int mrow  = mw * 16 + (lane & 15);
    const int ncol  = nw * 32 + (lane & 15);

    const int kSteps = K / BK;
    int buf = 0;
    for (int ks = 0; ks < kSteps; ++ks) {
        // prefetch next k-tile into registers (overlaps with WMMA on current tile)
        if (ks + 1 < kSteps) {
            aReg = *(const float4*)(Aptr + (long)(ks + 1) * BK);
            bReg = *(const float4*)(Bptr + (long)(ks + 1) * BK * N);
        }
        const float* Ab = As[buf];
        const float* Bb = Bs[buf];
#pragma unroll
        for (int kk = 0; kk < 4; ++kk) {
            const int kb = kk * 4 + koff;
            v2f a = *(const v2f*)&Ab[mrow * A_PAD + kb];   // ds_load_b64, 8B aligned
            v2f b0, b1;
            b0.x = Bb[(kb + 0) * B_PAD + ncol];
            b0.y = Bb[(kb + 1) * B_PAD + ncol];
            b1.x = Bb[(kb + 0) * B_PAD + ncol + 16];
            b1.y = Bb[(kb + 1) * B_PAD + ncol + 16];
            acc0 = __builtin_amdgcn_wmma_f32_16x16x4_f32(false, a, false, b0, (short)0, acc0, false, false);
            acc1 = __builtin_amdgcn_wmma_f32_16x16x4_f32(false, a, false, b1, (short)0, acc1, false, false);
        }
        __syncthreads();
        if (ks + 1 < kSteps) {
            buf ^= 1;
            *(float4*)&As[buf][am * A_PAD + ak] = aReg;
            *(float4*)&Bs[buf][bk * B_PAD + bn] = bReg;
            __syncthreads();
        }
    }

    // epilogue: C/D layout — VGPR r: lanes 0-15 -> M=r, lanes 16-31 -> M=r+8; N = lane&15
    const int rowBase = m0 + mw * 16 + (lane >> 4) * 8;
    const int c0 = n0 + ncol;
    const float bia0 = bias[c0];
    const float bia1 = bias[c0 + 16];
#pragma unroll
    for (int r = 0; r < 8; ++r) {
        float v0 = acc0[r] + bia0;
        float v1 = acc1[r] + bia1;
        if (ACT == 1) { v0 = fmaxf(v0, 0.f); v1 = fmaxf(v1, 0.f); }
        if (ACT == 2) { v0 = v0 > 0.f ? v0 : v0 * 0.001f; v1 = v1 > 0.f ? v1 : v1 * 0.001f; }
        const long row = rowBase + r;
        C[row * (long)ldC + c0]      = v0;
        C[row * (long)ldC + c0 + 16] = v1;
    }
}

__global__ void zero_kernel(float* __restrict__ p, int n)
{
    int i = blockIdx.x * blockDim.x + threadIdx.x;
    if (i < n) p[i] = 0.f;
}

// Column sum / sum-of-squares partials, atomically accumulated (coalesced: lanes -> consecutive cols)
__global__ __launch_bounds__(256) void bn_stats(
    const float* __restrict__ X, float* __restrict__ sum, float* __restrict__ sq,
    int cols, int rowsPer)
{
    const int col = blockIdx.x * blockDim.x + threadIdx.x;
    const long r0 = (long)blockIdx.y * rowsPer;
    float s = 0.f, q = 0.f;
    for (int r = 0; r < rowsPer; ++r) {
        float v = X[(r0 + r) * cols + col];
        s += v; q = fmaf(v, v, q);
    }
    atomicAdd(&sum[col], s);
    atomicAdd(&sq[col], q);
}

// Fused BN-normalize + per-sample routed head MLP. 1 wave (32 lanes) per sample, 8 samples/block.
__global__ __launch_bounds__(256) void head_kernel(
    const float* __restrict__ X, const float* __restrict__ sum, const float* __restrict__ sq,
    const float* __restrict__ gamma, const float* __restrict__ beta,
    const int* __restrict__ tt,
    const float* __restrict__ Wh1, const float* __restrict__ bh1,
    const float* __restrict__ Wh2, const float* __restrict__ bh2,
    const float* __restrict__ Wh3, const float* __restrict__ bh3,
    float* __restrict__ out)
{
    __shared__ float xn[8][2048];
    __shared__ float y1s[8][32];
    __shared__ float y2s[8][16];

    const int lane = threadIdx.x & 31;
    const int w    = threadIdx.x >> 5;
    const long b   = (long)blockIdx.x * 8 + w;
    const float invB = 1.0f / 8192.0f;

    // batchnorm normalize into LDS
    for (int i = lane; i < 2048; i += 32) {
        float m   = sum[i] * invB;
        float var = fmaf(-m, m, sq[i] * invB);
        float v   = (X[b * 2048 + i] - m) * rsqrtf(var + 1e-5f);
        xn[w][i]  = fmaf(v, gamma[i], beta[i]);
    }
    const int t = tt[b];
    __syncthreads();

    const float* W1 = Wh1 + (long)t * 2048 * 28;
    if (lane < 28) {
        float acc = bh1[t * 28 + lane];
        for (int f = 0; f < 2048; ++f) acc = fmaf(xn[w][f], W1[(long)f * 28 + lane], acc);
        y1s[w][lane] = fmaxf(acc, 0.f);
    }
    __syncthreads();

    const float* W2 = Wh2 + (long)t * 28 * 14;
    if (lane < 14) {
        float acc = bh2[t * 14 + lane];
#pragma unroll
        for (int f = 0; f < 28; ++f) acc = fmaf(y1s[w][f], W2[f * 14 + lane], acc);
        y2s[w][lane] = fmaxf(acc, 0.f);
    }
    __syncthreads();

    const float* W3 = Wh3 + (long)t * 14 * 5;
    if (lane < 5) {
        float acc = bh3[t * 5 + lane];
#pragma unroll
        for (int f = 0; f < 14; ++f) acc = fmaf(y2s[w][f], W3[f * 5 + lane], acc);
        out[b * 5 + lane] = acc;
    }
}

extern "C" void kernel_launch(void* const* d_in, const int* in_sizes, int n_in,
                              void* d_out, int out_size, void* d_ws, size_t ws_size,
                              hipStream_t stream) {
    const float* x_s   = (const float*)d_in[0];
    const float* x_p   = (const float*)d_in[1];
    const int*   tt    = (const int*)  d_in[2];
    const int*   dom   = (const int*)  d_in[3];   // device scalar domain_id
    const float* W_in  = (const float*)d_in[4];
    const float* b_in  = (const float*)d_in[5];
    const float* W_hid = (const float*)d_in[6];
    const float* b_hid = (const float*)d_in[7];
    const float* W_out = (const float*)d_in[8];
    const float* b_out = (const float*)d_in[9];
    const float* Wp    = (const float*)d_in[10];
    const float* bp    = (const float*)d_in[11];
    const float* gamma = (const float*)d_in[12];
    const float* beta  = (const float*)d_in[13];
    const float* Wh1   = (const float*)d_in[14];
    const float* bh1   = (const float*)d_in[15];
    const float* Wh2   = (const float*)d_in[16];
    const float* bh2   = (const float*)d_in[17];
    const float* Wh3   = (const float*)d_in[18];
    const float* bh3   = (const float*)d_in[19];
    float* out = (float*)d_out;

    const int Bn = 8192, F = 12288, SU = 4096, HID = 1024, X2 = 2048;

    // workspace layout (floats): bufA | bufB | stats(4096). xcat aliases bufB (free after last hidden layer).
    float* bufA  = (float*)d_ws;
    float* bufB  = bufA + (size_t)Bn * SU;
    float* stats = bufB + (size_t)Bn * SU;
    float* xcat  = bufB;

    dim3 blk(256);

    // shared branch: input layer
    gemm_wmma_f32<1><<<dim3(SU / BN, Bn / BM), blk, 0, stream>>>(
        x_s, W_in, b_in, bufA, Bn, SU, F, SU, nullptr, 0, 0);

    // 4 hidden layers, ping-pong
    float* src = bufA; float* dst = bufB;
    for (int l = 0; l < 4; ++l) {
        gemm_wmma_f32<1><<<dim3(SU / BN, Bn / BM), blk, 0, stream>>>(
            src, W_hid, b_hid, dst, Bn, SU, SU, SU, nullptr, 0, 0);
        float* t = src; src = dst; dst = t;
    }
    // final hidden activations are in src == bufA; bufB is now free -> reuse as xcat

    // shared output layer -> xcat[:, 0:1024]
    gemm_wmma_f32<1><<<dim3(HID / BN, Bn / BM), blk, 0, stream>>>(
        src, W_out, b_out, xcat, Bn, HID, SU, X2, nullptr, 0, 0);

    // private branch (device-side domain select) -> xcat[:, 1024:2048], leaky relu
    gemm_wmma_f32<2><<<dim3(HID / BN, Bn / BM), blk, 0, stream>>>(
        x_p, Wp, bp, xcat + HID, Bn, HID, F, X2, dom, (long)F * HID, (long)HID);

    // batchnorm statistics (zero accumulators every call: ws is not re-poisoned between replays)
    zero_kernel<<<(2 * X2 + 255) / 256, 256, 0, stream>>>(stats, 2 * X2);
    bn_stats<<<dim3(X2 / 256, 64), blk, 0, stream>>>(xcat, stats, stats + X2, X2, Bn / 64);

    // fused BN-normalize + routed heads
    head_kernel<<<Bn / 8, blk, 0, stream>>>(
        xcat, stats, stats + X2, gamma, beta, tt,
        Wh1, bh1, Wh2, bh2, Wh3, bh3, out);

    (void)in_sizes; (void)n_in; (void)out_size; (void)ws_size;
}